// Temp_FAM_GNNLayer_4389456576799
// MI455X (gfx1250) — compile-verified
//
#include <hip/hip_runtime.h>

// Problem constants (match reference)
#define NNODES 50000
#define NEDGES 800000
#define BDIM   2
#define INF    64
#define OUTF   64
#define NREL   8
#define NTYP   4
#define MROWS  (NNODES * BDIM)   // 100000 rows of length 64

typedef __attribute__((ext_vector_type(2))) float v2f;
typedef __attribute__((ext_vector_type(8))) float v8f;

// ---------------------------------------------------------------------------
// Kernel 1: x_all[n][r][b][o] = feat[n][b][:] @ weight[r][:][o] + m_bias[r][o]
// GEMM via V_WMMA_F32_16X16X4_F32. One wave handles a 16-row M tile for all
// 8 relations x 4 output column tiles. 100000 rows / 16 = 6250 tiles exactly.
// ---------------------------------------------------------------------------
__global__ __launch_bounds__(256)
void xall_wmma(const float* __restrict__ feat,     // [MROWS][64]
               const float* __restrict__ weight,   // [NREL][64][64]
               const float* __restrict__ m_bias,   // [NREL][64]
               float* __restrict__ xall)           // [NNODES][NREL][BDIM][64]
{
    const int lane = threadIdx.x & 31;
    const int wave = threadIdx.x >> 5;
    const int tile = blockIdx.x * 8 + wave;        // wave-uniform
    if (tile >= MROWS / 16) return;                // uniform exit: EXEC stays full

    const int half = lane >> 4;                    // 0: lanes 0-15, 1: lanes 16-31
    const int nsub = lane & 15;
    const int row0 = tile * 16;

    // ---- Load A tile (16 rows x 64 K) into WMMA layout: a[k] covers K=4k..4k+3
    // lane half h: a[k].x = feat[row][4k+2h], a[k].y = feat[row][4k+2h+1]
    const float* frow = feat + (size_t)(row0 + nsub) * 64;
    v2f a[16];
#pragma unroll
    for (int k = 0; k < 16; ++k)
        a[k] = *reinterpret_cast<const v2f*>(frow + 4 * k + 2 * half);

#pragma unroll 1
    for (int r = 0; r < NREL; ++r) {
        const float* wr = weight + (size_t)r * 64 * 64;   // [K=64][O=64]
#pragma unroll 1
        for (int t = 0; t < 4; ++t) {
            const int col = t * 16 + nsub;
            // ---- Load B tile (64 K x 16 cols) into WMMA layout:
            // b[k].x = W[4k+2h][col], b[k].y = W[4k+2h+1][col]
            v2f b[16];
#pragma unroll
            for (int k = 0; k < 16; ++k) {
                const float* wk = wr + (size_t)(4 * k + 2 * half) * 64 + col;
                b[k].x = wk[0];
                b[k].y = wk[64];
            }
            // ---- 16-step WMMA chain: D(16x16) += A(16x4) * B(4x16)
            v8f acc = {};
#pragma unroll
            for (int k = 0; k < 16; ++k)
                acc = __builtin_amdgcn_wmma_f32_16x16x4_f32(
                    false, a[k], false, b[k], (short)0, acc, false, false);

            const float bias = m_bias[r * 64 + col];
            // ---- Store: acc[j] = D[row j + 8*half][col]
#pragma unroll
            for (int j = 0; j < 8; ++j) {
                const int g  = row0 + j + 8 * half;        // global row = n*B + b
                const int n  = g >> 1;
                const int bi = g & 1;
                xall[(((size_t)n * NREL + r) * BDIM + bi) * 64 + col] = acc[j] + bias;
            }
        }
    }
}

// ---------------------------------------------------------------------------
// Kernel 2: out[n][b][o] = feat[n][b][:] @ loop_weight[ntype[n]][:][o]
//                          + h_bias[ntype[n]][o]
// One 128-thread block per node; feat row staged in LDS. Writes ALL of d_out.
// ---------------------------------------------------------------------------
__global__ __launch_bounds__(128)
void self_loop(const float* __restrict__ feat,      // [N][B][64]
               const float* __restrict__ loop_w,    // [NTYP][64][64]
               const float* __restrict__ h_bias,    // [NTYP][1][64]
               const int* __restrict__ ntypes,      // [N]
               float* __restrict__ out)             // [N][B][64]
{
    __shared__ float sf[BDIM * 64];
    const int n   = blockIdx.x;
    const int tid = threadIdx.x;

    sf[tid] = feat[(size_t)n * (BDIM * 64) + tid];
    __syncthreads();

    const int nt = ntypes[n];
    const int b  = tid >> 6;
    const int o  = tid & 63;

    const float* w = loop_w + (size_t)nt * 64 * 64 + o;
    const float* f = sf + b * 64;
    float acc = h_bias[nt * 64 + o];
#pragma unroll
    for (int i = 0; i < 64; ++i)
        acc = fmaf(f[i], w[(size_t)i * 64], acc);

    out[(size_t)n * (BDIM * 64) + tid] = acc;
}

// ---------------------------------------------------------------------------
// Kernel 3: per edge, gather 512B row from x_all[src,etype], scale by
// attention, atomic-add into out[dst]. One wave per edge, float4 per lane.
// ---------------------------------------------------------------------------
__global__ __launch_bounds__(256)
void edge_scatter(const float* __restrict__ xall,   // [N][R][B][64]
                  const float* __restrict__ att,    // [E]
                  const int* __restrict__ src,
                  const int* __restrict__ dst,
                  const int* __restrict__ etyp,
                  float* __restrict__ out,          // [N][B][64]
                  int nedges)
{
    const int lane = threadIdx.x & 31;
    const int wave = threadIdx.x >> 5;
    const int e    = blockIdx.x * 8 + wave;          // wave-uniform
    if (e >= nedges) return;

    const int   s = src[e];
    const int   d = dst[e];
    const int   r = etyp[e];
    const float aV = att[e];

    const float4* xrow = reinterpret_cast<const float4*>(
        xall + ((size_t)s * NREL + r) * (BDIM * 64));
    float4 v = xrow[lane];

    float* orow = out + (size_t)d * (BDIM * 64) + lane * 4;
    atomicAdd(orow + 0, aV * v.x);
    atomicAdd(orow + 1, aV * v.y);
    atomicAdd(orow + 2, aV * v.z);
    atomicAdd(orow + 3, aV * v.w);
}

// ---------------------------------------------------------------------------
extern "C" void kernel_launch(void* const* d_in, const int* in_sizes, int n_in,
                              void* d_out, int out_size, void* d_ws, size_t ws_size,
                              hipStream_t stream) {
    const float* feat    = (const float*)d_in[0];
    const float* weight  = (const float*)d_in[1];
    const float* m_bias  = (const float*)d_in[2];
    const float* loop_w  = (const float*)d_in[3];
    const float* h_bias  = (const float*)d_in[4];
    const float* att     = (const float*)d_in[5];
    const int*   src     = (const int*)d_in[6];
    const int*   dst     = (const int*)d_in[7];
    const int*   etyp    = (const int*)d_in[8];
    const int*   ntypes  = (const int*)d_in[9];
    float*       out     = (float*)d_out;
    float*       xall    = (float*)d_ws;   // needs N*R*B*64*4 = 204.8 MB

    const int nedges = in_sizes[6];

    // 1) x_all GEMM (WMMA). 6250 M-tiles, 8 waves (tiles) per 256-thread block.
    const int ntiles = MROWS / 16;
    xall_wmma<<<(ntiles + 7) / 8, 256, 0, stream>>>(feat, weight, m_bias, xall);

    // 2) self-loop path: fully writes d_out (stream-ordered before atomics).
    self_loop<<<NNODES, 128, 0, stream>>>(feat, loop_w, h_bias, ntypes, out);

    // 3) edge gather + attention scale + scatter-add.
    edge_scatter<<<(nedges + 7) / 8, 256, 0, stream>>>(xall, att, src, dst, etyp,
                                                       out, nedges);
}